// SSIMLoss_8744553414892
// MI455X (gfx1250) — compile-verified
//
#include <hip/hip_runtime.h>
#include <hip/hip_bf16.h>

typedef float v2f __attribute__((ext_vector_type(2)));
typedef float v8f __attribute__((ext_vector_type(8)));

#define IMG_H 512
#define IMG_W 512
#define PLANES 48            // 16 batch * 3 channels
#define OUT_TH 16            // output tile rows per workgroup
#define OUT_TW 64            // output tile cols per workgroup
#define PATCH_H 32           // 16+10 halo, padded to 2 rowgroups of 16
#define PATCH_W 76           // 64+10 halo, padded so A can read col cbase+27 (max 75)
#define H_PITCH 68           // padded pitch for intermediate (bank-conflict dodge)
#define H_ROWS 32
#define NPIX 12582912.0      // 16*3*512*512
#define C1 1.0e-4f
#define C2 9.0e-4f

// normalized gaussian, sigma=1.5, K=11
__device__ const float GK[11] = {
    0.001028362f, 0.007598758f, 0.036000773f, 0.109360697f, 0.213005537f,
    0.266011729f,
    0.213005537f, 0.109360697f, 0.036000773f, 0.007598758f, 0.001028362f};

__device__ __forceinline__ float gval(int d) {
    return ((unsigned)d < 11u) ? GK[d] : 0.0f;
}

__device__ __forceinline__ v8f wmma_f32(v2f a, v2f b, v8f c) {
    return __builtin_amdgcn_wmma_f32_16x16x4_f32(
        /*neg_a=*/false, a, /*neg_b=*/false, b,
        /*c_mod=*/(short)0, c, /*reuse_a=*/false, /*reuse_b=*/false);
}

__global__ __launch_bounds__(128) void ssim_tile_kernel(
    const float* __restrict__ pred, const float* __restrict__ targ,
    float* __restrict__ partial) {
    __shared__ float sP[PATCH_H * PATCH_W];
    __shared__ float sT[PATCH_H * PATCH_W];
    __shared__ float sH[5][H_ROWS * H_PITCH];
    __shared__ float swsum[4];

    const int tid  = threadIdx.x;
    const int lane = tid & 31;
    const int wave = tid >> 5;
    const int lh   = lane >> 4;   // half-wave select (0/1)
    const int l15  = lane & 15;

    const int tx = blockIdx.x;    // 0..7   (x tiles of 64)
    const int ty = blockIdx.y;    // 0..31  (y tiles of 16)
    const int pl = blockIdx.z;    // 0..47  (N*C planes)

    const float* __restrict__ P = pred + (size_t)pl * IMG_H * IMG_W;
    const float* __restrict__ T = targ + (size_t)pl * IMG_H * IMG_W;
    const int gy0 = ty * OUT_TH - 5;
    const int gx0 = tx * OUT_TW - 5;

    // ---- cooperative global -> LDS patch load, zero-padded (SAME conv) ----
    for (int i = tid; i < PATCH_H * PATCH_W; i += 128) {
        int r = i / PATCH_W;
        int c = i - r * PATCH_W;
        int gy = gy0 + r, gx = gx0 + c;
        bool inb = ((unsigned)gy < (unsigned)IMG_H) && ((unsigned)gx < (unsigned)IMG_W);
        int off = gy * IMG_W + gx;
        sP[i] = inb ? P[off] : 0.0f;
        sT[i] = inb ? T[off] : 0.0f;
    }

    // ---- banded weight fragments (serve as B in h-pass, A in v-pass) ----
    // frag[k]: lane<16 holds W rows 4k,4k+1 at col l15; lane>=16 rows 4k+2,4k+3
    v2f wf[7];
#pragma unroll
    for (int k = 0; k < 7; ++k) {
        int d = 4 * k + 2 * lh - l15;
        wf[k].x = gval(d);
        wf[k].y = gval(d + 1);
    }

    __syncthreads();

    // ---- horizontal pass: H(rows x 16cols) = Patch(rows x K) * Wh(K x 16) ----
    // 8 jobs = 2 rowgroups x 4 colgroups; wave w does jobs w and w+4.
    for (int j = wave; j < 8; j += 4) {
        const int rg = j >> 2, cg = j & 3;
        const int rbase = rg * 16, cbase = cg * 16;
        v8f hp = {}, ht = {}, hpp = {}, htt = {}, hpt = {};
#pragma unroll
        for (int k = 0; k < 7; ++k) {
            int pi = (rbase + l15) * PATCH_W + cbase + 4 * k + 2 * lh;
            v2f ap, at;
            ap.x = sP[pi]; ap.y = sP[pi + 1];
            at.x = sT[pi]; at.y = sT[pi + 1];
            v2f app = ap * ap, att = at * at, apt = ap * at;
            v2f b = wf[k];
            hp  = wmma_f32(ap,  b, hp);
            ht  = wmma_f32(at,  b, ht);
            hpp = wmma_f32(app, b, hpp);
            htt = wmma_f32(att, b, htt);
            hpt = wmma_f32(apt, b, hpt);
        }
        // store D tiles: VGPR r -> row r (lanes<16) / row r+8 (lanes>=16)
        const int hr = rbase + 8 * lh;
        const int hc = cbase + l15;
#pragma unroll
        for (int r = 0; r < 8; ++r) {
            int idx = (hr + r) * H_PITCH + hc;
            sH[0][idx] = hp[r];
            sH[1][idx] = ht[r];
            sH[2][idx] = hpp[r];
            sH[3][idx] = htt[r];
            sH[4][idx] = hpt[r];
        }
    }

    __syncthreads();

    // ---- vertical pass: Out(16x16) = Wv(16 x K) * H(K x 16); wave w -> colgroup w
    {
        const int cg = wave;
        v8f vp = {}, vt = {}, vpp = {}, vtt = {}, vpt = {};
#pragma unroll
        for (int k = 0; k < 7; ++k) {
            v2f a = wf[k];
            int i0 = (4 * k + 2 * lh) * H_PITCH + cg * 16 + l15;
            int i1 = i0 + H_PITCH;
            v2f bp, bt, bpp, btt, bpt;
            bp.x  = sH[0][i0]; bp.y  = sH[0][i1];
            bt.x  = sH[1][i0]; bt.y  = sH[1][i1];
            bpp.x = sH[2][i0]; bpp.y = sH[2][i1];
            btt.x = sH[3][i0]; btt.y = sH[3][i1];
            bpt.x = sH[4][i0]; bpt.y = sH[4][i1];
            vp  = wmma_f32(a, bp,  vp);
            vt  = wmma_f32(a, bt,  vt);
            vpp = wmma_f32(a, bpp, vpp);
            vtt = wmma_f32(a, btt, vtt);
            vpt = wmma_f32(a, bpt, vpt);
        }

        // ---- SSIM formula per output pixel, sum within lane ----
        float loc = 0.0f;
#pragma unroll
        for (int r = 0; r < 8; ++r) {
            float mu1 = vp[r], mu2 = vt[r];
            float mu1sq = mu1 * mu1, mu2sq = mu2 * mu2, mu12 = mu1 * mu2;
            float s1  = vpp[r] - mu1sq;
            float s2  = vtt[r] - mu2sq;
            float s12 = vpt[r] - mu12;
            float num = (2.0f * mu12 + C1) * (2.0f * s12 + C2);
            float den = (mu1sq + mu2sq + C1) * (s1 + s2 + C2);
            loc += num / den;
        }
        // wave32 reduce
#pragma unroll
        for (int o = 16; o > 0; o >>= 1)
            loc += __shfl_down(loc, o, 32);
        if (lane == 0) swsum[wave] = loc;
    }

    __syncthreads();
    if (tid == 0) {
        float s = swsum[0] + swsum[1] + swsum[2] + swsum[3];
        partial[((size_t)blockIdx.z * gridDim.y + blockIdx.y) * gridDim.x + blockIdx.x] = s;
    }
}

__global__ __launch_bounds__(256) void ssim_reduce_kernel(
    const float* __restrict__ partial, float* __restrict__ out, int n) {
    __shared__ double sd[256];
    double a = 0.0;
    for (int i = threadIdx.x; i < n; i += 256) a += (double)partial[i];
    sd[threadIdx.x] = a;
    __syncthreads();
    for (int s = 128; s > 0; s >>= 1) {
        if (threadIdx.x < s) sd[threadIdx.x] += sd[threadIdx.x + s];
        __syncthreads();
    }
    if (threadIdx.x == 0) out[0] = 1.0f - (float)(sd[0] / NPIX);
}

extern "C" void kernel_launch(void* const* d_in, const int* in_sizes, int n_in,
                              void* d_out, int out_size, void* d_ws, size_t ws_size,
                              hipStream_t stream) {
    const float* pred = (const float*)d_in[0];
    const float* targ = (const float*)d_in[1];
    float* partial = (float*)d_ws;           // 8*32*48 = 12288 floats (48 KB)
    float* out = (float*)d_out;

    dim3 grid(IMG_W / OUT_TW, IMG_H / OUT_TH, PLANES);  // (8, 32, 48)
    ssim_tile_kernel<<<grid, 128, 0, stream>>>(pred, targ, partial);

    int nblocks = (IMG_W / OUT_TW) * (IMG_H / OUT_TH) * PLANES;  // 12288
    ssim_reduce_kernel<<<1, 256, 0, stream>>>(partial, out, nblocks);
}